// QuantizerEMA_22488448762078
// MI455X (gfx1250) — compile-verified
//
#include <hip/hip_runtime.h>
#include <hip/hip_bf16.h>

// ---------------------------------------------------------------------------
// VQ-VAE EMA quantizer for MI455X (gfx1250, wave32, WMMA).
// N = 65536 rows, D = 256 dims, K = 1024 codes.
// Distance GEMM runs on V_WMMA_F32_16X16X4_F32 with 4 interleaved
// accumulator chains (4 N-tiles per K-sweep) to keep the matrix pipe full
// and amortize the A-fragment LDS load 4x.
// ---------------------------------------------------------------------------

typedef float v2f __attribute__((ext_vector_type(2)));
typedef float v8f __attribute__((ext_vector_type(8)));

#define N_ROWS  65536
#define DIM     256
#define NCODE   1024
#define BM      128          // rows per block (8 waves x 16 rows)
#define DPAD    258          // LDS row stride in floats (bank-conflict padding)
#define DECAY   0.99f
#define EPS     1e-5f
#define COMMIT  0.25f

// ---- workspace layout (in floats) ----
#define WS_ENORM   0                    // 1024
#define WS_NCN     1024                 // 1024
#define WS_CNT     2048                 // 1024
#define WS_LOSS    3072                 // 1    (cnt..loss zeroed as one range)
#define WS_DW      4096                 // 262144
#define WS_CLOSEST (4096 + 262144)      // 65536 ints

// ---- output layout (in floats) ----
#define OUT_LOSS 0
#define OUT_ST   1
#define OUT_IDX  (1 + N_ROWS * DIM)
#define OUT_EMB  (OUT_IDX + N_ROWS)
#define OUT_CLU  (OUT_EMB + NCODE * DIM)
#define OUT_EMA  (OUT_CLU + NCODE)

__global__ void vq_zero_kernel(float* __restrict__ p, int n) {
    int i = blockIdx.x * blockDim.x + threadIdx.x;
    for (; i < n; i += gridDim.x * blockDim.x) p[i] = 0.0f;
}

// one block (256 threads) per code: enorm[c] = sum_d emb[c][d]^2
__global__ void vq_enorm_kernel(const float* __restrict__ emb,
                                float* __restrict__ enorm) {
    __shared__ float red[256];
    int c = blockIdx.x, d = threadIdx.x;
    float v = emb[c * DIM + d];
    red[d] = v * v;
    __syncthreads();
    for (int s = 128; s > 0; s >>= 1) {
        if (d < s) red[d] += red[d + s];
        __syncthreads();
    }
    if (d == 0) enorm[c] = red[0];
}

// ---------------------------------------------------------------------------
// Distance + argmin via V_WMMA_F32_16X16X4_F32, 4-way accumulator interleave.
// Block: 256 threads = 8 waves; wave w owns rows [base + 16w, base + 16w + 16).
// ---------------------------------------------------------------------------
__global__ __launch_bounds__(256)
void vq_argmin_wmma_kernel(const float* __restrict__ z,
                           const float* __restrict__ emb,
                           const float* __restrict__ enorm,
                           int* __restrict__ closest) {
    __shared__ float zs[BM][DPAD];   // ~132 KB of the 320 KB WGP LDS
    __shared__ float znl[BM];
    __shared__ float enl[NCODE];

    const int tid  = threadIdx.x;
    const int lane = tid & 31;
    const int w    = tid >> 5;       // wave id 0..7
    const int half = lane >> 4;      // 0: K pair {0,1} / rows M,  1: K pair {2,3} / rows M+8
    const int l16  = lane & 15;
    const long rowBase = (long)blockIdx.x * BM;

    // --- stage z tile (coalesced float4), plus enorm table ---
    const float4* z4 = (const float4*)(z + rowBase * DIM);
    for (int i = tid; i < BM * (DIM / 4); i += 256) {
        int r = i >> 6, c = i & 63;
        float4 v = z4[r * 64 + c];
        zs[r][c * 4 + 0] = v.x;
        zs[r][c * 4 + 1] = v.y;
        zs[r][c * 4 + 2] = v.z;
        zs[r][c * 4 + 3] = v.w;
    }
    for (int i = tid; i < NCODE; i += 256) enl[i] = enorm[i];
    __syncthreads();

    // --- per-row squared norms ---
    if (tid < BM) {
        float s = 0.0f;
        for (int d = 0; d < DIM; ++d) { float v = zs[tid][d]; s += v * v; }
        znl[tid] = s;
    }
    __syncthreads();

    // loop-invariant row norms for this wave's C-layout rows
    float zn[8];
#pragma unroll
    for (int r = 0; r < 8; ++r) zn[r] = znl[w * 16 + r + 8 * half];

    float bestd[8];
    int   besti[8];
#pragma unroll
    for (int r = 0; r < 8; ++r) { bestd[r] = 3.402823466e38f; besti[r] = 0; }

    const float* zrow = &zs[w * 16 + l16][2 * half];

    for (int n0 = 0; n0 < NCODE; n0 += 64) {
        const float* eb0 = emb + (long)(n0 + l16) * DIM + 2 * half;
        const float* eb1 = eb0 + 16 * DIM;
        const float* eb2 = eb0 + 32 * DIM;
        const float* eb3 = eb0 + 48 * DIM;
        if (n0 + 64 < NCODE)   // pre-pull next 64-code block
            __builtin_prefetch(eb0 + 64 * DIM, 0, 3);

        v8f acc0 = {0.f, 0.f, 0.f, 0.f, 0.f, 0.f, 0.f, 0.f};
        v8f acc1 = acc0, acc2 = acc0, acc3 = acc0;

#pragma unroll 4
        for (int k0 = 0; k0 < DIM; k0 += 4) {
            v2f a  = *(const v2f*)(zrow + k0);   // shared A: 16x4 f32 fragment
            v2f b0 = *(const v2f*)(eb0 + k0);    // B: 4x16 f32 fragments (= e^T)
            v2f b1 = *(const v2f*)(eb1 + k0);
            v2f b2 = *(const v2f*)(eb2 + k0);
            v2f b3 = *(const v2f*)(eb3 + k0);
            // 4 independent accumulator chains keep the matrix pipe busy
            acc0 = __builtin_amdgcn_wmma_f32_16x16x4_f32(false, a, false, b0,
                                                         (short)0, acc0, false, false);
            acc1 = __builtin_amdgcn_wmma_f32_16x16x4_f32(false, a, false, b1,
                                                         (short)0, acc1, false, false);
            acc2 = __builtin_amdgcn_wmma_f32_16x16x4_f32(false, a, false, b2,
                                                         (short)0, acc2, false, false);
            acc3 = __builtin_amdgcn_wmma_f32_16x16x4_f32(false, a, false, b3,
                                                         (short)0, acc3, false, false);
        }

#pragma unroll
        for (int j = 0; j < 4; ++j) {
            const v8f acc = (j == 0) ? acc0 : (j == 1) ? acc1 : (j == 2) ? acc2 : acc3;
            const float en = enl[n0 + 16 * j + l16];
            const int   ni = n0 + 16 * j + l16;
#pragma unroll
            for (int r = 0; r < 8; ++r) {
                // C/D layout: VGPR r -> rows r (lanes 0-15) and r+8 (lanes 16-31)
                float d = zn[r] + en - 2.0f * acc[r];
                if (d < bestd[r]) { bestd[r] = d; besti[r] = ni; }
            }
        }
    }

    // --- min-reduce across the 16 lanes of each half (ties -> smaller index) ---
#pragma unroll
    for (int r = 0; r < 8; ++r) {
        float bd = bestd[r];
        int   bi = besti[r];
#pragma unroll
        for (int off = 8; off >= 1; off >>= 1) {
            float od = __shfl_xor(bd, off, 32);
            int   oi = __shfl_xor(bi, off, 32);
            if (od < bd || (od == bd && oi < bi)) { bd = od; bi = oi; }
        }
        if (l16 == 0)
            closest[rowBase + w * 16 + 8 * half + r] = bi;
    }
}

// ---------------------------------------------------------------------------
// Block-per-row: gather quantized (== st output), scatter dw/counts,
// accumulate commitment-loss partial.
// ---------------------------------------------------------------------------
__global__ void vq_gather_scatter_kernel(const float* __restrict__ z,
                                         const float* __restrict__ emb,
                                         const int* __restrict__ closest,
                                         float* __restrict__ out_st,
                                         float* __restrict__ out_idx,
                                         float* __restrict__ dw,
                                         float* __restrict__ cnt,
                                         float* __restrict__ losssum) {
    __shared__ float red[256];
    const int row = blockIdx.x;
    const int d   = threadIdx.x;
    const int idx = closest[row];
    const long zi = (long)row * DIM + d;

    float zv = z[zi];
    float q  = emb[(long)idx * DIM + d];
    out_st[zi] = q;                        // z + sg(q - z) == q numerically
    atomicAdd(&dw[(long)idx * DIM + d], zv);
    if (d == 0) {
        atomicAdd(&cnt[idx], 1.0f);
        out_idx[row] = (float)idx;
    }
    float diff = q - zv;
    red[d] = diff * diff;
    __syncthreads();
    for (int s = 128; s > 0; s >>= 1) {
        if (d < s) red[d] += red[d + s];
        __syncthreads();
    }
    if (d == 0) atomicAdd(losssum, red[0]);
}

// single block, 1024 threads: EMA cluster sizes + Laplace norm + loss finalize
__global__ void vq_cluster_kernel(const float* __restrict__ cs,
                                  const float* __restrict__ cnt,
                                  const float* __restrict__ losssum,
                                  float* __restrict__ out_loss,
                                  float* __restrict__ out_clu,
                                  float* __restrict__ ncn) {
    __shared__ float red[NCODE];
    int t = threadIdx.x;
    float nc = cs[t] * DECAY + cnt[t] * (1.0f - DECAY);
    red[t] = nc;
    __syncthreads();
    for (int s = 512; s > 0; s >>= 1) {
        if (t < s) red[t] += red[t + s];
        __syncthreads();
    }
    float n = red[0];
    float v = (nc + EPS) / (n + (float)NCODE * EPS) * n;
    out_clu[t] = v;
    ncn[t] = v;
    if (t == 0)
        out_loss[0] = losssum[0] / ((float)N_ROWS * (float)DIM) * COMMIT;
}

// elementwise: new_ema and new_embeddings = new_ema / new_cluster[code]
__global__ void vq_embed_kernel(const float* __restrict__ ema,
                                const float* __restrict__ dw,
                                const float* __restrict__ ncn,
                                float* __restrict__ out_emb,
                                float* __restrict__ out_ema) {
    int i = blockIdx.x * blockDim.x + threadIdx.x;   // 0 .. 262143
    float e = ema[i] * DECAY + dw[i] * (1.0f - DECAY);
    out_ema[i] = e;
    out_emb[i] = e / ncn[i >> 8];
}

extern "C" void kernel_launch(void* const* d_in, const int* in_sizes, int n_in,
                              void* d_out, int out_size, void* d_ws, size_t ws_size,
                              hipStream_t stream) {
    const float* z    = (const float*)d_in[0];   // [65536, 256]
    const float* emb  = (const float*)d_in[1];   // [1024, 256]
    const float* ema  = (const float*)d_in[2];   // [1024, 256]
    const float* csiz = (const float*)d_in[3];   // [1024]

    float* out = (float*)d_out;
    float* ws  = (float*)d_ws;

    float* enorm   = ws + WS_ENORM;
    float* ncn     = ws + WS_NCN;
    float* cnt     = ws + WS_CNT;
    float* losssum = ws + WS_LOSS;
    float* dw      = ws + WS_DW;
    int*   closest = (int*)(ws + WS_CLOSEST);

    // 1) zero scratch: cnt + loss (contiguous 1025), dw (262144)
    vq_zero_kernel<<<8, 256, 0, stream>>>(cnt, NCODE + 1);
    vq_zero_kernel<<<1024, 256, 0, stream>>>(dw, NCODE * DIM);

    // 2) codebook squared norms
    vq_enorm_kernel<<<NCODE, 256, 0, stream>>>(emb, enorm);

    // 3) f32 WMMA distance GEMM + row argmin (4-way accumulator interleave)
    vq_argmin_wmma_kernel<<<N_ROWS / BM, 256, 0, stream>>>(z, emb, enorm, closest);

    // 4) gather + scatter + loss partials
    vq_gather_scatter_kernel<<<N_ROWS, 256, 0, stream>>>(
        z, emb, closest, out + OUT_ST, out + OUT_IDX, dw, cnt, losssum);

    // 5) cluster EMA / normalization / loss
    vq_cluster_kernel<<<1, NCODE, 0, stream>>>(
        csiz, cnt, losssum, out + OUT_LOSS, out + OUT_CLU, ncn);

    // 6) new ema + new embeddings
    vq_embed_kernel<<<(NCODE * DIM) / 256, 256, 0, stream>>>(
        ema, dw, ncn, out + OUT_EMB, out + OUT_EMA);
}